// S4Block_73461120631470
// MI455X (gfx1250) — compile-verified
//
#include <hip/hip_runtime.h>
#include <hip/hip_bf16.h>

// ---------------------------------------------------------------------------
// Mamba-style S4 block for MI455X (gfx1250), wave32.
// fp32 WMMA (V_WMMA_F32_16X16X4_F32) GEMMs + async-to-LDS double-buffered
// staging (GLOBAL_LOAD_ASYNC_TO_LDS_B128, ASYNCcnt) + fused selective scan.
//   D_MODEL=1024, D_INNER=2048, D_STATE=16, D_CONV=4, B=2, L=2048
// ---------------------------------------------------------------------------

#define D_MODEL 1024
#define D_INNER 2048
#define D_STATE 16
#define D_CONV  4
#define BATCH   2
#define SEQLEN  2048
#define MROWS   (BATCH * SEQLEN)          // 4096

typedef __attribute__((ext_vector_type(2))) float v2f;
typedef __attribute__((ext_vector_type(8))) float v8f;

// ---------------- CDNA5 async-to-LDS helpers (inline asm, portable) --------
// VDST VGPR holds the wave-relative LDS byte offset; for a generic pointer to
// a __shared__ object that is exactly the low 32 address bits (LDS aperture).
__device__ __forceinline__ unsigned lds_off(const void* p) {
    return (unsigned)(unsigned long long)p;
}
__device__ __forceinline__ void async_ld_b128(unsigned lds, const void* g) {
    asm volatile("global_load_async_to_lds_b128 %0, %1, off"
                 :: "v"(lds), "v"(g) : "memory");
}
__device__ __forceinline__ void async_ld_b32(unsigned lds, const void* g) {
    asm volatile("global_load_async_to_lds_b32 %0, %1, off"
                 :: "v"(lds), "v"(g) : "memory");
}
__device__ __forceinline__ void wait_async0() {
    asm volatile("s_wait_asynccnt 0x0" ::: "memory");
}

// ---------------------------------------------------------------------------
// Kernel 1/5: C[M x N] = A[M x K] @ B[N x K]^T   (fp32 WMMA)
// Workgroup tile 128x64, 8 waves (4x2), each wave a 32x32 (2x2 of 16x16) tile.
// K staged in chunks of 16 through DOUBLE-BUFFERED LDS via async-to-LDS;
// one barrier per chunk.  LDS row stride 20 floats: 20*lane mod 64 hits 16
// distinct banks and keeps 8B/16B alignment.
// Requires M%128==0, N%64==0, K%16==0 (true for all uses here).
// ---------------------------------------------------------------------------
#define TM 128
#define TN 64
#define KB 16
#define KST 20   // padded LDS row stride (floats)

__global__ __launch_bounds__(256) void gemm_nt_wmma_f32(
    const float* __restrict__ A, const float* __restrict__ Bm,
    float* __restrict__ C, int M, int N, int K)
{
    __shared__ float As[2][TM * KST];   // 2 x 10240 B
    __shared__ float Bs[2][TN * KST];   // 2 x  5120 B

    const int tid  = threadIdx.x;
    const int lane = tid & 31;
    const int wave = tid >> 5;            // 0..7
    const int wm   = wave >> 1;           // 0..3  (M direction)
    const int wn   = wave & 1;            // 0..1  (N direction)
    const int row0 = blockIdx.y * TM;
    const int col0 = blockIdx.x * TN;
    const int half = lane >> 4;           // 0/1
    const int lr   = lane & 15;

    // staging assignment: A tile = 128x16 f32 = 512 float4 (2/thread),
    //                     B tile =  64x16 f32 = 256 float4 (1/thread)
    const int rA0 = tid >> 2;                   // 0..63
    const int rA1 = (tid + 256) >> 2;           // 64..127
    const int cA  = (tid & 3) * 4;              // 0,4,8,12
    const int rB  = tid >> 2;
    const float* Ap0 = A  + (size_t)(row0 + rA0) * K + cA;
    const float* Ap1 = A  + (size_t)(row0 + rA1) * K + cA;
    const float* Bp  = Bm + (size_t)(col0 + rB ) * K + cA;

    const unsigned lA0[2] = { lds_off(&As[0][rA0 * KST + cA]),
                              lds_off(&As[1][rA0 * KST + cA]) };
    const unsigned lA1[2] = { lds_off(&As[0][rA1 * KST + cA]),
                              lds_off(&As[1][rA1 * KST + cA]) };
    const unsigned lB [2] = { lds_off(&Bs[0][rB  * KST + cA]),
                              lds_off(&Bs[1][rB  * KST + cA]) };

    v8f acc[2][2] = {};                   // 2x2 of 16x16 f32 tiles

    const int nc = K / KB;
    // prologue: stage chunk 0 into buffer 0
    async_ld_b128(lA0[0], Ap0);
    async_ld_b128(lA1[0], Ap1);
    async_ld_b128(lB [0], Bp );
    wait_async0();

    for (int c = 0; c < nc; ++c) {
        __syncthreads();                  // publish chunk c; fences buf reuse
        const int buf = c & 1;
        if (c + 1 < nc) {                 // prefetch chunk c+1 into buf^1
            const int k0 = (c + 1) * KB;
            async_ld_b128(lA0[buf ^ 1], Ap0 + k0);
            async_ld_b128(lA1[buf ^ 1], Ap1 + k0);
            async_ld_b128(lB [buf ^ 1], Bp  + k0);
        }
        // ---- 4 K-steps of 4, 4 WMMA each ----
        #pragma unroll
        for (int kk = 0; kk < KB; kk += 4) {
            v2f a[2], b[2];
            // A 16x4 fragment: VGPR0 = K0 (lanes 0-15) / K2 (lanes 16-31),
            //                  VGPR1 = K1 / K3  -> contiguous f32 pair.
            #pragma unroll
            for (int i = 0; i < 2; ++i) {
                int m = wm * 32 + i * 16 + lr;
                a[i] = *reinterpret_cast<const v2f*>(
                    &As[buf][m * KST + kk + 2 * half]);
            }
            // B 4x16 fragment (N across lanes, K across VGPRs), same pairing.
            #pragma unroll
            for (int j = 0; j < 2; ++j) {
                int n = wn * 32 + j * 16 + lr;
                b[j] = *reinterpret_cast<const v2f*>(
                    &Bs[buf][n * KST + kk + 2 * half]);
            }
            #pragma unroll
            for (int i = 0; i < 2; ++i)
                #pragma unroll
                for (int j = 0; j < 2; ++j)
                    acc[i][j] = __builtin_amdgcn_wmma_f32_16x16x4_f32(
                        false, a[i], false, b[j], (short)0, acc[i][j],
                        false, false);
        }
        if (c + 1 < nc) wait_async0();    // chunk c+1 resident before barrier
    }

    // ---- write back: VGPR v -> row (v + 8*half), col = lr ----
    #pragma unroll
    for (int i = 0; i < 2; ++i)
        #pragma unroll
        for (int j = 0; j < 2; ++j)
            #pragma unroll
            for (int v = 0; v < 8; ++v) {
                int r = row0 + wm * 32 + i * 16 + v + 8 * half;
                int c = col0 + wn * 32 + j * 16 + lr;
                C[(size_t)r * N + c] = acc[i][j][v];
            }
}

// ---------------------------------------------------------------------------
// Kernel 2/5: causal depthwise conv (4 taps) + bias + SiLU.
// x_ssm is the first D_INNER columns of xz (row-major rows of 2*D_INNER).
// ---------------------------------------------------------------------------
__global__ __launch_bounds__(256) void conv_silu_kernel(
    const float* __restrict__ xz, const float* __restrict__ cw,
    const float* __restrict__ cb, float* __restrict__ xact)
{
    int idx = blockIdx.x * 256 + threadIdx.x;   // b*L*D_INNER + l*D_INNER + d
    int d = idx & (D_INNER - 1);
    int l = (idx >> 11) & (SEQLEN - 1);
    int b = idx >> 22;
    float s = cb[d];
    #pragma unroll
    for (int j = 0; j < D_CONV; ++j) {
        int ll = l - (D_CONV - 1) + j;
        if (ll >= 0)
            s += cw[d * D_CONV + j] *
                 xz[((size_t)(b * SEQLEN + ll)) * (2 * D_INNER) + d];
    }
    xact[idx] = s / (1.f + __expf(-s));         // SiLU
}

// ---------------------------------------------------------------------------
// Kernel 3/5: proj[row, n] = dot(x_act[row,:], W_x[n,:])  n in 0..32.
// One wave32 per (row, n); lane-strided dot + shfl_xor reduction.
// ---------------------------------------------------------------------------
__global__ __launch_bounds__(256) void proj_kernel(
    const float* __restrict__ xact, const float* __restrict__ Wx,
    float* __restrict__ proj)
{
    int gw   = (blockIdx.x * 256 + threadIdx.x) >> 5;   // global wave id
    int lane = threadIdx.x & 31;
    int row  = gw / (2 * D_STATE + 1);
    int n    = gw % (2 * D_STATE + 1);
    const float* xr = xact + (size_t)row * D_INNER;
    const float* wr = Wx   + (size_t)n   * D_INNER;
    float acc = 0.f;
    for (int k = lane; k < D_INNER; k += 32)
        acc += xr[k] * wr[k];
    #pragma unroll
    for (int off = 16; off > 0; off >>= 1)
        acc += __shfl_xor(acc, off, 32);
    if (lane == 0)
        proj[(size_t)row * (2 * D_STATE + 1) + n] = acc;
}

// ---------------------------------------------------------------------------
// Kernel 4/5: fused selective scan + skip + z-gate.
// One thread per (b, d): h[16] and A[16] live in registers.  Wave 0
// async-prefetches the next step's 32-float [B_t | C_t] row into a ping-pong
// LDS buffer one step ahead (ASYNCcnt), one barrier per timestep.
// ---------------------------------------------------------------------------
__global__ __launch_bounds__(256) void scan_kernel(
    const float* __restrict__ xz,   const float* __restrict__ xact,
    const float* __restrict__ proj, const float* __restrict__ Alog,
    const float* __restrict__ Dp,   const float* __restrict__ Wdt,
    const float* __restrict__ bdt,  float* __restrict__ yact)
{
    const int d = blockIdx.x * 256 + threadIdx.x;   // 0..D_INNER-1
    const int b = blockIdx.y;

    __shared__ float BC[2][2 * D_STATE];   // [buf][ B(16) | C(16) ]

    float Areg[D_STATE];
    #pragma unroll
    for (int n = 0; n < D_STATE; ++n)
        Areg[n] = -__expf(Alog[d * D_STATE + n]);   // A = -exp(A_log)

    float h[D_STATE];
    #pragma unroll
    for (int n = 0; n < D_STATE; ++n) h[n] = 0.f;

    const float wdt = Wdt[d];
    const float bd  = bdt[d];
    const float Dv  = Dp[d];
    const int   PW  = 2 * D_STATE + 1;              // 33

    // prologue: wave 0 stages [B|C] of step 0 (one async b32 per lane)
    if (threadIdx.x < 32) {
        const float* p0 = proj + (size_t)(b * SEQLEN) * PW + threadIdx.x;
        async_ld_b32(lds_off(&BC[0][threadIdx.x]), p0);
        wait_async0();
    }

    for (int l = 0; l < SEQLEN; ++l) {
        __syncthreads();                            // publish BC[l&1]
        const int cur = l & 1;
        if (threadIdx.x < 32 && l + 1 < SEQLEN) {   // prefetch step l+1
            const float* pn =
                proj + (size_t)(b * SEQLEN + l + 1) * PW + threadIdx.x;
            async_ld_b32(lds_off(&BC[cur ^ 1][threadIdx.x]), pn);
        }

        const size_t row = (size_t)(b * SEQLEN + l);
        float pl = proj[row * PW + 2 * D_STATE];    // uniform -> scalar load
        float t  = pl * wdt + bd;
        float dt = (t > 20.f) ? t : __logf(1.f + __expf(t));    // softplus
        float x  = xact[row * D_INNER + d];
        float xdt = x * dt;
        float y = 0.f;
        #pragma unroll
        for (int n = 0; n < D_STATE; ++n) {
            float dA = __expf(dt * Areg[n]);
            h[n] = dA * h[n] + xdt * BC[cur][n];
            y   += h[n] * BC[cur][D_STATE + n];
        }
        y += x * Dv;
        float zv = xz[row * (2 * D_INNER) + D_INNER + d];
        y *= zv / (1.f + __expf(-zv));                          // y * silu(z)
        yact[row * D_INNER + d] = y;

        if (threadIdx.x < 32 && l + 1 < SEQLEN) wait_async0();
    }
}

// ---------------------------------------------------------------------------
// Launch
// ---------------------------------------------------------------------------
extern "C" void kernel_launch(void* const* d_in, const int* in_sizes, int n_in,
                              void* d_out, int out_size, void* d_ws, size_t ws_size,
                              hipStream_t stream)
{
    const float* x      = (const float*)d_in[0];
    const float* W_in   = (const float*)d_in[1];
    const float* conv_w = (const float*)d_in[2];
    const float* conv_b = (const float*)d_in[3];
    const float* W_x    = (const float*)d_in[4];
    const float* A_log  = (const float*)d_in[5];
    const float* D_par  = (const float*)d_in[6];
    const float* W_dt   = (const float*)d_in[7];
    const float* b_dt   = (const float*)d_in[8];
    const float* W_out  = (const float*)d_in[9];
    float* out = (float*)d_out;

    // workspace layout (floats)
    float* ws   = (float*)d_ws;
    float* XZ   = ws;                                        // 4096 x 4096
    float* XACT = XZ   + (size_t)MROWS * 2 * D_INNER;        // 4096 x 2048
    float* PROJ = XACT + (size_t)MROWS * D_INNER;            // 4096 x 33
    float* YACT = PROJ + (size_t)MROWS * (2 * D_STATE + 1);  // 4096 x 2048

    // 1) xz = x @ W_in^T          M=4096 N=4096 K=1024
    gemm_nt_wmma_f32<<<dim3((2 * D_INNER) / TN, MROWS / TM), 256, 0, stream>>>(
        x, W_in, XZ, MROWS, 2 * D_INNER, D_MODEL);

    // 2) causal conv + SiLU
    conv_silu_kernel<<<(BATCH * SEQLEN * D_INNER) / 256, 256, 0, stream>>>(
        XZ, conv_w, conv_b, XACT);

    // 3) proj = x_act @ W_x^T     (4096 rows x 33 cols), one wave per output
    {
        int waves  = MROWS * (2 * D_STATE + 1);   // 135168
        int blocks = waves / 8;                   // 8 waves / block, exact
        proj_kernel<<<blocks, 256, 0, stream>>>(XACT, W_x, PROJ);
    }

    // 4) fused selective scan + skip + z-gate
    scan_kernel<<<dim3(D_INNER / 256, BATCH), 256, 0, stream>>>(
        XZ, XACT, PROJ, A_log, D_par, W_dt, b_dt, YACT);

    // 5) out = y @ W_out^T        M=4096 N=1024 K=2048
    gemm_nt_wmma_f32<<<dim3(D_MODEL / TN, MROWS / TM), 256, 0, stream>>>(
        YACT, W_out, out, MROWS, D_MODEL, D_INNER);
}